// SGCwithJK_40020505264511
// MI455X (gfx1250) — compile-verified
//
#include <hip/hip_runtime.h>

typedef float v2f __attribute__((ext_vector_type(2)));
typedef float v8f __attribute__((ext_vector_type(8)));

#define FD 128   // feature dim

// ---------------- degree / normalization ----------------

__global__ void deg_init_kernel(float* deg, int N) {
  int i = blockIdx.x * blockDim.x + threadIdx.x;
  if (i < N) deg[i] = 1.0f;                 // self loop contributes 1
}

__global__ void deg_accum_kernel(const int* __restrict__ col, float* deg, int E) {
  int e = blockIdx.x * blockDim.x + threadIdx.x;
  if (e < E) atomicAdd(&deg[col[e]], 1.0f);
}

__global__ void dinv_kernel(float* deg, int N) {
  int i = blockIdx.x * blockDim.x + threadIdx.x;
  if (i < N) deg[i] = rsqrtf(deg[i]);       // deg >= 1 always (self loops)
}

// ---------------- propagate ----------------
// dst[i,:] = dinv[i]^2 * src[i,:]   (self-loop term; also initializes dst)
__global__ void selfloop_kernel(const float* __restrict__ src, const float* __restrict__ dinv,
                                float* __restrict__ dst, int N) {
  int t = blockIdx.x * blockDim.x + threadIdx.x;
  int node = t >> 5, lane = t & 31;
  if (node >= N) return;
  float w = dinv[node]; w *= w;
  float4 v = ((const float4*)(src + (size_t)node * FD))[lane];
  ((float4*)(dst + (size_t)node * FD))[lane] =
      make_float4(w * v.x, w * v.y, w * v.z, w * v.w);
}

// one wave32 per edge: dst[col,:] += dinv[row]*dinv[col] * src[row,:]
// row/col/dinv loads are wave-uniform; feature loads are coalesced float4;
// accumulation via native global_atomic_add_f32 (L2-resident: h fits in 192MB L2).
__global__ void edge_prop_kernel(const int* __restrict__ row, const int* __restrict__ col,
                                 const float* __restrict__ dinv,
                                 const float* __restrict__ src, float* __restrict__ dst, int E) {
  int t = blockIdx.x * blockDim.x + threadIdx.x;
  int e = t >> 5, lane = t & 31;
  if (e >= E) return;
  int r = row[e], c = col[e];
  float w = dinv[r] * dinv[c];
  float4 v = ((const float4*)(src + (size_t)r * FD))[lane];
  float* d = dst + (size_t)c * FD + lane * 4;
  atomicAdd(d + 0, w * v.x);
  atomicAdd(d + 1, w * v.y);
  atomicAdd(d + 2, w * v.z);
  atomicAdd(d + 3, w * v.w);
}

// ---------------- dense GEMM via fp32 WMMA ----------------
// Block = 256 threads = 8 waves; block computes a 16-row x 128-col output strip.
// Wave w owns the 16x16 tile at columns [16w, 16w+16). K loop: 32 x wmma_f32_16x16x4_f32.
// A-frag (16x4 f32): lanes 0-15 -> M=lane, K={k0,k0+1}; lanes 16-31 -> K={k0+2,k0+3}.
// B-frag (4x16 f32): lanes give N, vgpr pair gives K (mirror of A).
// C/D (16x16 f32): vgpr i -> M = i + 8*(lane>>4), N = lane&15.
// mode 0: out = relu(A@W + b)        mode 1: out = max(h1, h2, A@W + b)   (fused JK)
__global__ __launch_bounds__(256)
void gemm_wmma_kernel(const float* __restrict__ A, const float* __restrict__ W,
                      const float* __restrict__ bias,
                      const float* __restrict__ h1, const float* __restrict__ h2,
                      float* __restrict__ out, int N, int mode) {
  const int wave  = threadIdx.x >> 5;
  const int lane  = threadIdx.x & 31;
  const int hi    = lane >> 4;     // 0 or 1 (selects K pair)
  const int lm    = lane & 15;
  const int mbase = blockIdx.x * 16;
  const int ncol  = wave * 16 + lm;

  int arow = mbase + lm;
  if (arow >= N) arow = N - 1;     // clamp instead of exiting: EXEC must stay all-1s for WMMA
  const float* ap = A + (size_t)arow * FD + 2 * hi;
  const float* wp = W + (size_t)(2 * hi) * FD + ncol;

  v8f acc = {};
#pragma unroll
  for (int k0 = 0; k0 < FD; k0 += 4) {
    v2f a;
    a.x = ap[k0];
    a.y = ap[k0 + 1];
    v2f b;
    b.x = wp[(size_t)k0 * FD];
    b.y = wp[(size_t)(k0 + 1) * FD];
    acc = __builtin_amdgcn_wmma_f32_16x16x4_f32(false, a, false, b,
                                                (short)0, acc, false, false);
  }

  float bv = bias[ncol];
#pragma unroll
  for (int i = 0; i < 8; ++i) {
    int r = mbase + i + 8 * hi;
    if (r >= N) continue;
    size_t idx = (size_t)r * FD + ncol;
    float v = acc[i] + bv;
    if (mode == 0) out[idx] = fmaxf(v, 0.0f);
    else           out[idx] = fmaxf(fmaxf(h1[idx], h2[idx]), v);
  }
}

// ---------------- driver ----------------

extern "C" void kernel_launch(void* const* d_in, const int* in_sizes, int n_in,
                              void* d_out, int out_size, void* d_ws, size_t ws_size,
                              hipStream_t stream) {
  const float* x  = (const float*)d_in[0];
  const int*   ei = (const int*)d_in[1];
  const float* W1 = (const float*)d_in[2];
  const float* b1 = (const float*)d_in[3];
  const float* W2 = (const float*)d_in[4];
  const float* b2 = (const float*)d_in[5];
  const float* W3 = (const float*)d_in[6];
  const float* b3 = (const float*)d_in[7];
  float* out = (float*)d_out;

  const int N = in_sizes[0] / FD;
  const int E = in_sizes[1] / 2;
  const int* row = ei;          // edge_index[0]
  const int* col = ei + E;      // edge_index[1]

  // workspace: dinv (padded to 51200 floats) | p | h1 | h2
  float* ws   = (float*)d_ws;
  float* dinv = ws;
  size_t nd   = (size_t)N * FD;
  float* p    = ws + 51200;
  float* h1   = p + nd;
  float* h2   = h1 + nd;

  const int TB = 256;
  const int nBlocks = (N + TB - 1) / TB;
  const int eBlocks = (E + TB - 1) / TB;

  // gcn_norm: deg = 1 + indeg, dinv = rsqrt(deg)
  deg_init_kernel<<<nBlocks, TB, 0, stream>>>(dinv, N);
  deg_accum_kernel<<<eBlocks, TB, 0, stream>>>(col, dinv, E);
  dinv_kernel<<<nBlocks, TB, 0, stream>>>(dinv, N);

  const long nodeWork = (long)N * 32;
  const long edgeWork = (long)E * 32;
  const int nodeBlocks = (int)((nodeWork + TB - 1) / TB);
  const int edgeBlocks = (int)((edgeWork + TB - 1) / TB);
  const int gemmBlocks = (N + 15) / 16;

  // layer 1: h1 = relu(P(x) @ W1 + b1)
  selfloop_kernel<<<nodeBlocks, TB, 0, stream>>>(x, dinv, p, N);
  edge_prop_kernel<<<edgeBlocks, TB, 0, stream>>>(row, col, dinv, x, p, E);
  gemm_wmma_kernel<<<gemmBlocks, TB, 0, stream>>>(p, W1, b1, nullptr, nullptr, h1, N, 0);

  // layer 2: h2 = relu(P(h1) @ W2 + b2)
  selfloop_kernel<<<nodeBlocks, TB, 0, stream>>>(h1, dinv, p, N);
  edge_prop_kernel<<<edgeBlocks, TB, 0, stream>>>(row, col, dinv, h1, p, E);
  gemm_wmma_kernel<<<gemmBlocks, TB, 0, stream>>>(p, W2, b2, nullptr, nullptr, h2, N, 0);

  // layer 3 + JumpingKnowledge max, fused into the GEMM epilogue
  selfloop_kernel<<<nodeBlocks, TB, 0, stream>>>(h2, dinv, p, N);
  edge_prop_kernel<<<edgeBlocks, TB, 0, stream>>>(row, col, dinv, h2, p, E);
  gemm_wmma_kernel<<<gemmBlocks, TB, 0, stream>>>(p, W3, b3, h1, h2, out, N, 1);
}